// LSTMDecoder_20298015441420
// MI455X (gfx1250) — compile-verified
//
#include <hip/hip_runtime.h>
#include <hip/hip_bf16.h>
#include <math.h>

// Problem constants (from the reference)
#define Bb 64
#define Ll 196
#define Ee 2048
#define Dd 512
#define Aa 512
#define Mm 512
#define Vv 30000
#define Tt 21
#define MLen 20   // Tt - 1

typedef __attribute__((ext_vector_type(2))) float v2f;
typedef __attribute__((ext_vector_type(8))) float v8f;

// ---------------------------------------------------------------------------
// WMMA fragment layouts (V_WMMA_F32_16X16X4_F32, ISA 7.12.2):
//   A 16x4 : lanes 0-15 hold M=lane, v0=K0,v1=K1 ; lanes 16-31 same M, K2,K3
//   B 4x16 : lanes 0-15 hold N=lane, v0=K0,v1=K1 ; lanes 16-31 same N, K2,K3
//   C 16x16: vgpr r -> (M=r, N=lane) lanes 0-15 ; (M=r+8, N=lane-16) lanes 16-31
// ---------------------------------------------------------------------------

// ============================ packed-B GEMM =================================
// Wp layout: for col tile ct (16 cols), k-chunk ks (4 ks), lane (32):
//   Wp[(ct*(K/4) + ks)*32 + lane] = { W[ks*4+half*2][ct*16+l16],
//                                     W[ks*4+half*2+1][ct*16+l16] }
// so a wave's B fragment for one WMMA is ONE contiguous global_load_b64 and
// the pointer advances by an immediate 256B per k-step.
// ============================================================================
__global__ __launch_bounds__(256)
void pack_w_kernel(const float* __restrict__ W, v2f* __restrict__ Wp,
                   int K, int N)
{
  const size_t total = (size_t)(N / 16) * (K / 4) * 32;
  const size_t stride = (size_t)gridDim.x * blockDim.x;
  for (size_t i = (size_t)blockIdx.x * blockDim.x + threadIdx.x; i < total;
       i += stride) {
    const int lane = (int)(i & 31);
    const size_t j = i >> 5;
    const int ks = (int)(j % (size_t)(K / 4));
    const int ct = (int)(j / (size_t)(K / 4));
    const int l16 = lane & 15, half = lane >> 4;
    const int k = ks * 4 + half * 2;
    const int n = ct * 16 + l16;
    v2f v;
    v.x = W[(size_t)k * N + n];
    v.y = W[(size_t)(k + 1) * N + n];
    Wp[i] = v;
  }
}

__device__ __forceinline__ void wmma_stripe_packed(
    const float* __restrict__ A, int lda,
    const v2f* __restrict__ Wt,   // pre-offset to this col tile
    int K, int row0, int l16, int half, int lane, v8f (&acc)[4])
{
  const int KS = K >> 2;
  const float* a[4];
  #pragma unroll
  for (int rt = 0; rt < 4; ++rt)
    a[rt] = A + (size_t)(row0 + rt * 16 + l16) * lda + half * 2;
  const v2f* bp = Wt + lane;

  // 2-stage software pipeline: fragments for step ks+1 load while the four
  // WMMAs of step ks issue -> loads overlap math instead of loadcnt==0 stalls.
  v2f bc = bp[0];
  v2f ac[4];
  #pragma unroll
  for (int rt = 0; rt < 4; ++rt) ac[rt] = *(const v2f*)(a[rt]);

  #pragma unroll 2
  for (int ks = 0; ks < KS - 1; ++ks) {
    v2f bn = bp[(size_t)(ks + 1) * 32];
    v2f an[4];
    #pragma unroll
    for (int rt = 0; rt < 4; ++rt)
      an[rt] = *(const v2f*)(a[rt] + (size_t)(ks + 1) * 4);
    #pragma unroll
    for (int rt = 0; rt < 4; ++rt)
      acc[rt] = __builtin_amdgcn_wmma_f32_16x16x4_f32(
          false, ac[rt], false, bc, (short)0, acc[rt], false, false);
    bc = bn;
    #pragma unroll
    for (int rt = 0; rt < 4; ++rt) ac[rt] = an[rt];
  }
  #pragma unroll
  for (int rt = 0; rt < 4; ++rt)
    acc[rt] = __builtin_amdgcn_wmma_f32_16x16x4_f32(
        false, ac[rt], false, bc, (short)0, acc[rt], false, false);
}

__global__ __launch_bounds__(128)
void wmma_gemm_f32_packed(const float* __restrict__ A,  int lda,
                          const v2f*  __restrict__ Wp,
                          const float* __restrict__ A2, int lda2,
                          const v2f*  __restrict__ Wp2,
                          const float* __restrict__ bias,
                          float* __restrict__ C, int ldc,
                          int N, int K, int K2, int act,
                          const int* __restrict__ cap_lens, int t)
{
  const int lane = threadIdx.x & 31;
  const int wave = threadIdx.x >> 5;
  const int l16  = lane & 15;
  const int half = lane >> 4;
  const int ct   = blockIdx.x * 4 + wave;     // 16-col tile index
  const int col0 = ct * 16;
  const int row0 = blockIdx.y * 64;
  if (col0 >= N) return;  // wave-uniform: EXEC stays all-ones around WMMA

  v8f acc[4] = {};
  wmma_stripe_packed(A, lda, Wp + (size_t)ct * (K >> 2) * 32,
                     K, row0, l16, half, lane, acc);
  if (A2)
    wmma_stripe_packed(A2, lda2, Wp2 + (size_t)ct * (K2 >> 2) * 32,
                       K2, row0, l16, half, lane, acc);

  const int n = col0 + l16;
  const float bv = bias ? bias[n] : 0.0f;
  #pragma unroll
  for (int rt = 0; rt < 4; ++rt) {
    #pragma unroll
    for (int r = 0; r < 8; ++r) {
      const int m = row0 + rt * 16 + half * 8 + r;
      float v = acc[rt][r] + bv;
      if (act == 1)      v = tanhf(v);
      else if (act == 2) v = 1.0f / (1.0f + expf(-v));
      if (cap_lens)      v = ((cap_lens[m] - 1) > t) ? v : 0.0f;
      C[(size_t)m * ldc + n] = v;
    }
  }
}

// ===================== unpacked GEMM (ws_size fallback) =====================
__device__ __forceinline__ void wmma_stripe_f32(
    const float* __restrict__ A, int lda,
    const float* __restrict__ W, int ldw,
    int K, int row0, int col0, int l16, int half, v8f (&acc)[4])
{
  const float* a0 = A + (size_t)(row0 + l16) * lda;
  const float* wc = W + col0 + l16;
  #pragma unroll 4
  for (int k = 0; k < K; k += 4) {
    const int ka = k + half * 2;
    v2f bv;
    bv.x = wc[(size_t)ka * ldw];
    bv.y = wc[(size_t)ka * ldw + ldw];
    #pragma unroll
    for (int rt = 0; rt < 4; ++rt) {
      const float* ar = a0 + (size_t)(rt * 16) * lda;
      v2f av;
      av.x = ar[ka];
      av.y = ar[ka + 1];
      acc[rt] = __builtin_amdgcn_wmma_f32_16x16x4_f32(
          false, av, false, bv, (short)0, acc[rt], false, false);
    }
  }
}

__global__ __launch_bounds__(128)
void wmma_gemm_f32(const float* __restrict__ A,  int lda,
                   const float* __restrict__ W,  int ldw,
                   const float* __restrict__ A2, int lda2,
                   const float* __restrict__ W2, int ldw2,
                   const float* __restrict__ bias,
                   float* __restrict__ C, int ldc,
                   int N, int K, int K2, int act,
                   const int* __restrict__ cap_lens, int t)
{
  const int lane = threadIdx.x & 31;
  const int wave = threadIdx.x >> 5;
  const int l16  = lane & 15;
  const int half = lane >> 4;
  const int col0 = (blockIdx.x * 4 + wave) * 16;
  const int row0 = blockIdx.y * 64;
  if (col0 >= N) return;

  v8f acc[4] = {};
  wmma_stripe_f32(A, lda, W, ldw, K, row0, col0, l16, half, acc);
  if (A2) wmma_stripe_f32(A2, lda2, W2, ldw2, K2, row0, col0, l16, half, acc);

  const int n = col0 + l16;
  const float bv = bias ? bias[n] : 0.0f;
  #pragma unroll
  for (int rt = 0; rt < 4; ++rt) {
    #pragma unroll
    for (int r = 0; r < 8; ++r) {
      const int m = row0 + rt * 16 + half * 8 + r;
      float v = acc[rt][r] + bv;
      if (act == 1)      v = tanhf(v);
      else if (act == 2) v = 1.0f / (1.0f + expf(-v));
      if (cap_lens)      v = ((cap_lens[m] - 1) > t) ? v : 0.0f;
      C[(size_t)m * ldc + n] = v;
    }
  }
}

// ---------------------------------------------------------------------------
// mean over L:  mean[b,e] = (1/L) sum_l enc[b,l,e]
// ---------------------------------------------------------------------------
__global__ __launch_bounds__(256)
void mean_kernel(const float* __restrict__ enc, float* __restrict__ mean)
{
  const int idx = blockIdx.x * blockDim.x + threadIdx.x;   // B*E threads
  if (idx >= Bb * Ee) return;
  const int b = idx >> 11;
  const int e = idx & (Ee - 1);
  const float* p = enc + (size_t)b * Ll * Ee + e;
  float s = 0.0f;
  #pragma unroll 4
  for (int l = 0; l < Ll; ++l) s += p[(size_t)l * Ee];
  mean[idx] = s * (1.0f / (float)Ll);
}

__global__ __launch_bounds__(256)
void bias_sum_kernel(const float* __restrict__ b0, const float* __restrict__ b1,
                     float* __restrict__ o)
{
  const int i = blockIdx.x * blockDim.x + threadIdx.x;
  if (i < 4 * Dd) o[i] = b0[i] + b1[i];
}

// ---------------------------------------------------------------------------
// Attention + context + x-assembly. One block per batch row.
// ---------------------------------------------------------------------------
__global__ __launch_bounds__(256)
void attention_kernel(const float* __restrict__ enc,      // [B,L,E]
                      const float* __restrict__ e_proj,   // [B,L,A]
                      const float* __restrict__ d,        // [B,A]
                      const float* __restrict__ w_att,    // [A]
                      const float* __restrict__ b_att,    // [1]
                      const float* __restrict__ gate,     // [B,E]
                      const float* __restrict__ emb,      // [V,M]
                      const int*   __restrict__ captions, // [B,T]
                      const int*   __restrict__ cap_lens, // [B]
                      float* __restrict__ x,              // [B, M+E]
                      float* __restrict__ alphas_out,     // [B, MLen, L] base
                      int t)
{
  __shared__ float d_s[Aa];
  __shared__ float w_s[Aa];
  __shared__ float e_s[200];
  __shared__ float red[8];

  const int b    = blockIdx.x;
  const int tid  = threadIdx.x;
  const int lane = tid & 31;
  const int wave = tid >> 5;

  for (int k = tid; k < Aa; k += 256) { d_s[k] = d[b * Aa + k]; w_s[k] = w_att[k]; }
  __syncthreads();

  const float* ep = e_proj + (size_t)b * Ll * Aa;
  for (int l = wave; l < Ll; l += 8) {
    const float* row = ep + (size_t)l * Aa;
    float s = 0.0f;
    for (int k = lane; k < Aa; k += 32) {
      float v = row[k] + d_s[k];
      s += (v > 0.0f ? v : 0.0f) * w_s[k];
    }
    #pragma unroll
    for (int o = 16; o > 0; o >>= 1) s += __shfl_xor(s, o, 32);
    if (lane == 0) e_s[l] = s + b_att[0];
  }
  __syncthreads();

  const float v = (tid < Ll) ? e_s[tid] : -3.0e38f;
  float mx = v;
  #pragma unroll
  for (int o = 16; o > 0; o >>= 1) mx = fmaxf(mx, __shfl_xor(mx, o, 32));
  if (lane == 0) red[wave] = mx;
  __syncthreads();
  const float maxv = fmaxf(fmaxf(fmaxf(red[0], red[1]), fmaxf(red[2], red[3])),
                           fmaxf(fmaxf(red[4], red[5]), fmaxf(red[6], red[7])));
  __syncthreads();
  const float ex = (tid < Ll) ? expf(v - maxv) : 0.0f;
  float sm = ex;
  #pragma unroll
  for (int o = 16; o > 0; o >>= 1) sm += __shfl_xor(sm, o, 32);
  if (lane == 0) red[wave] = sm;
  __syncthreads();
  const float sumv = red[0] + red[1] + red[2] + red[3] +
                     red[4] + red[5] + red[6] + red[7];
  const float mf = ((cap_lens[b] - 1) > t) ? 1.0f : 0.0f;
  if (tid < Ll) {
    const float al = ex / sumv;
    e_s[tid] = al;
    alphas_out[(size_t)(b * MLen + t) * Ll + tid] = al * mf;
  }
  __syncthreads();

  const float* encb = enc + (size_t)b * Ll * Ee;
  float* xb = x + (size_t)b * (Mm + Ee);
  for (int e = tid; e < Ee; e += 256) {
    float s = 0.0f;
    for (int l = 0; l < Ll; ++l) s += e_s[l] * encb[(size_t)l * Ee + e];
    xb[Mm + e] = gate[b * Ee + e] * s;
  }
  const int tok = captions[b * Tt + t];
  for (int m = tid; m < Mm; m += 256) xb[m] = emb[(size_t)tok * Mm + m];
}

// ---------------------------------------------------------------------------
// LSTM cell elementwise
// ---------------------------------------------------------------------------
__global__ __launch_bounds__(256)
void lstm_cell_kernel(const float* __restrict__ gates,
                      float* __restrict__ h, float* __restrict__ c)
{
  const int idx = blockIdx.x * blockDim.x + threadIdx.x;   // B*D
  if (idx >= Bb * Dd) return;
  const int b = idx >> 9;
  const int j = idx & (Dd - 1);
  const float* g = gates + (size_t)b * 4 * Dd;
  const float gi = g[j], gf = g[Dd + j], gg = g[2 * Dd + j], go = g[3 * Dd + j];
  const float si = 1.0f / (1.0f + expf(-gi));
  const float sf = 1.0f / (1.0f + expf(-gf));
  const float so = 1.0f / (1.0f + expf(-go));
  const float cn = sf * c[idx] + si * tanhf(gg);
  c[idx] = cn;
  h[idx] = so * tanhf(cn);
}

// ---------------------------------------------------------------------------
// Host side
// ---------------------------------------------------------------------------
static inline void launch_gemm(hipStream_t s,
                               const float* A, int lda, const float* W, int ldw,
                               const float* A2, int lda2, const float* W2, int ldw2,
                               const float* bias, float* C, int ldc,
                               int M, int N, int K, int K2, int act,
                               const int* cap_lens, int t)
{
  dim3 grid((N + 63) / 64, M / 64);
  wmma_gemm_f32<<<grid, 128, 0, s>>>(A, lda, W, ldw, A2, lda2, W2, ldw2,
                                     bias, C, ldc, N, K, K2, act, cap_lens, t);
}

static inline void launch_gemm_packed(hipStream_t s,
                                      const float* A, int lda, const v2f* Wp,
                                      const float* A2, int lda2, const v2f* Wp2,
                                      const float* bias, float* C, int ldc,
                                      int M, int N, int K, int K2, int act,
                                      const int* cap_lens, int t)
{
  dim3 grid((N + 63) / 64, M / 64);
  wmma_gemm_f32_packed<<<grid, 128, 0, s>>>(A, lda, Wp, A2, lda2, Wp2,
                                            bias, C, ldc, N, K, K2, act,
                                            cap_lens, t);
}

static inline void launch_pack(hipStream_t s, const float* W, v2f* Wp,
                               int K, int N)
{
  size_t total = (size_t)(N / 16) * (K / 4) * 32;
  int blocks = (int)((total + 255) / 256);
  if (blocks > 2048) blocks = 2048;
  pack_w_kernel<<<blocks, 256, 0, s>>>(W, Wp, K, N);
}

extern "C" void kernel_launch(void* const* d_in, const int* in_sizes, int n_in,
                              void* d_out, int out_size, void* d_ws, size_t ws_size,
                              hipStream_t stream)
{
  const float* enc      = (const float*)d_in[0];   // [B,L,E]
  const int*   captions = (const int*)  d_in[1];   // [B,T]
  const int*   cap_lens = (const int*)  d_in[2];   // [B]
  const float* W_enc    = (const float*)d_in[3];   // [E,A]
  const float* b_enc    = (const float*)d_in[4];
  const float* W_dec    = (const float*)d_in[5];   // [D,A]
  const float* b_dec    = (const float*)d_in[6];
  const float* w_att    = (const float*)d_in[7];   // [A,1]
  const float* b_att    = (const float*)d_in[8];
  const float* emb      = (const float*)d_in[9];   // [V,M]
  const float* W_ih     = (const float*)d_in[10];  // [M+E,4D]
  const float* b_ih     = (const float*)d_in[11];
  const float* W_hh     = (const float*)d_in[12];  // [D,4D]
  const float* b_hh     = (const float*)d_in[13];
  const float* W_h0     = (const float*)d_in[14];  // [E,D]
  const float* b_h0     = (const float*)d_in[15];
  const float* W_c0     = (const float*)d_in[16];  // [E,D]
  const float* b_c0     = (const float*)d_in[17];
  const float* W_beta   = (const float*)d_in[18];  // [D,E]
  const float* b_beta   = (const float*)d_in[19];
  const float* W_fc     = (const float*)d_in[20];  // [D,V]
  const float* b_fc     = (const float*)d_in[21];

  float* out = (float*)d_out;
  const size_t PRED_OFF = (size_t)Bb * MLen * Vv;
  float* alphas_out = out + PRED_OFF;

  // ---- workspace layout (floats) ----
  float* ws = (float*)d_ws;
  size_t o = 0;
  float* e_proj = ws + o; o += (size_t)Bb * Ll * Aa;
  float* hbuf   = ws + o; o += (size_t)Bb * Dd;
  float* cbuf   = ws + o; o += (size_t)Bb * Dd;
  float* meanb  = ws + o; o += (size_t)Bb * Ee;
  float* dbuf   = ws + o; o += (size_t)Bb * Aa;
  float* gateb  = ws + o; o += (size_t)Bb * Ee;
  float* xbuf   = ws + o; o += (size_t)Bb * (Mm + Ee);
  float* gatesb = ws + o; o += (size_t)Bb * 4 * Dd;
  float* bcomb  = ws + o; o += 4 * Dd;
  const size_t base_floats = o;

  // packed weights (v2f each = 8B), appended after the base region
  v2f* pk = (v2f*)(ws + base_floats);
  size_t p = 0;
  v2f* P_enc  = pk + p; p += (size_t)(Aa / 16)     * (Ee / 4)        * 32;
  v2f* P_h0   = pk + p; p += (size_t)(Dd / 16)     * (Ee / 4)        * 32;
  v2f* P_c0   = pk + p; p += (size_t)(Dd / 16)     * (Ee / 4)        * 32;
  v2f* P_dec  = pk + p; p += (size_t)(Aa / 16)     * (Dd / 4)        * 32;
  v2f* P_beta = pk + p; p += (size_t)(Ee / 16)     * (Dd / 4)        * 32;
  v2f* P_ih   = pk + p; p += (size_t)(4 * Dd / 16) * ((Mm + Ee) / 4) * 32;
  v2f* P_hh   = pk + p; p += (size_t)(4 * Dd / 16) * (Dd / 4)        * 32;
  v2f* P_fc   = pk + p; p += (size_t)(Vv / 16)     * (Dd / 4)        * 32;
  const size_t need = base_floats * sizeof(float) + p * sizeof(v2f);
  const bool packed = (ws_size >= need);

  // ---- prelude ----
  mean_kernel<<<(Bb * Ee + 255) / 256, 256, 0, stream>>>(enc, meanb);
  bias_sum_kernel<<<(4 * Dd + 255) / 256, 256, 0, stream>>>(b_ih, b_hh, bcomb);

  if (packed) {
    launch_pack(stream, W_enc,  P_enc,  Ee,      Aa);
    launch_pack(stream, W_h0,   P_h0,   Ee,      Dd);
    launch_pack(stream, W_c0,   P_c0,   Ee,      Dd);
    launch_pack(stream, W_dec,  P_dec,  Dd,      Aa);
    launch_pack(stream, W_beta, P_beta, Dd,      Ee);
    launch_pack(stream, W_ih,   P_ih,   Mm + Ee, 4 * Dd);
    launch_pack(stream, W_hh,   P_hh,   Dd,      4 * Dd);
    launch_pack(stream, W_fc,   P_fc,   Dd,      Vv);

    launch_gemm_packed(stream, meanb, Ee, P_h0, nullptr, 0, nullptr,
                       b_h0, hbuf, Dd, Bb, Dd, Ee, 0, 1, nullptr, 0);
    launch_gemm_packed(stream, meanb, Ee, P_c0, nullptr, 0, nullptr,
                       b_c0, cbuf, Dd, Bb, Dd, Ee, 0, 1, nullptr, 0);
    launch_gemm_packed(stream, enc, Ee, P_enc, nullptr, 0, nullptr,
                       b_enc, e_proj, Aa, Bb * Ll, Aa, Ee, 0, 0, nullptr, 0);

    for (int t = 0; t < MLen; ++t) {
      launch_gemm_packed(stream, hbuf, Dd, P_dec, nullptr, 0, nullptr,
                         b_dec, dbuf, Aa, Bb, Aa, Dd, 0, 0, nullptr, 0);
      launch_gemm_packed(stream, hbuf, Dd, P_beta, nullptr, 0, nullptr,
                         b_beta, gateb, Ee, Bb, Ee, Dd, 0, 2, nullptr, 0);
      attention_kernel<<<Bb, 256, 0, stream>>>(enc, e_proj, dbuf, w_att, b_att,
                                               gateb, emb, captions, cap_lens,
                                               xbuf, alphas_out, t);
      launch_gemm_packed(stream, xbuf, Mm + Ee, P_ih, hbuf, Dd, P_hh,
                         bcomb, gatesb, 4 * Dd, Bb, 4 * Dd, Mm + Ee, Dd, 0,
                         nullptr, 0);
      lstm_cell_kernel<<<(Bb * Dd + 255) / 256, 256, 0, stream>>>(gatesb, hbuf,
                                                                  cbuf);
      launch_gemm_packed(stream, hbuf, Dd, P_fc, nullptr, 0, nullptr,
                         b_fc, out + (size_t)t * Vv, MLen * Vv, Bb, Vv, Dd, 0,
                         0, cap_lens, t);
    }
  } else {
    // fallback: unpacked-weight path (smaller workspace requirement)
    launch_gemm(stream, meanb, Ee, W_h0, Dd, nullptr, 0, nullptr, 0,
                b_h0, hbuf, Dd, Bb, Dd, Ee, 0, 1, nullptr, 0);
    launch_gemm(stream, meanb, Ee, W_c0, Dd, nullptr, 0, nullptr, 0,
                b_c0, cbuf, Dd, Bb, Dd, Ee, 0, 1, nullptr, 0);
    launch_gemm(stream, enc, Ee, W_enc, Aa, nullptr, 0, nullptr, 0,
                b_enc, e_proj, Aa, Bb * Ll, Aa, Ee, 0, 0, nullptr, 0);

    for (int t = 0; t < MLen; ++t) {
      launch_gemm(stream, hbuf, Dd, W_dec, Aa, nullptr, 0, nullptr, 0,
                  b_dec, dbuf, Aa, Bb, Aa, Dd, 0, 0, nullptr, 0);
      launch_gemm(stream, hbuf, Dd, W_beta, Ee, nullptr, 0, nullptr, 0,
                  b_beta, gateb, Ee, Bb, Ee, Dd, 0, 2, nullptr, 0);
      attention_kernel<<<Bb, 256, 0, stream>>>(enc, e_proj, dbuf, w_att, b_att,
                                               gateb, emb, captions, cap_lens,
                                               xbuf, alphas_out, t);
      launch_gemm(stream, xbuf, Mm + Ee, W_ih, 4 * Dd, hbuf, Dd, W_hh, 4 * Dd,
                  bcomb, gatesb, 4 * Dd, Bb, 4 * Dd, Mm + Ee, Dd, 0,
                  nullptr, 0);
      lstm_cell_kernel<<<(Bb * Dd + 255) / 256, 256, 0, stream>>>(gatesb, hbuf,
                                                                  cbuf);
      launch_gemm(stream, hbuf, Dd, W_fc, Vv, nullptr, 0, nullptr, 0,
                  b_fc, out + (size_t)t * Vv, MLen * Vv, Bb, Vv, Dd, 0, 0,
                  cap_lens, t);
    }
  }
}